// TFTapasComputeColumnLogits_55336358641993
// MI455X (gfx1250) — compile-verified
//
#include <hip/hip_runtime.h>
#include <hip/hip_bf16.h>

typedef __attribute__((ext_vector_type(2))) float v2f;
typedef __attribute__((ext_vector_type(4))) float v4f;
typedef __attribute__((ext_vector_type(8))) float v8f;

#define NUM_ROWS 64
#define NUM_COLS 32
#define NUM_CELLS (NUM_ROWS * NUM_COLS)
#define HIDDEN 1024
#define BATCH 64
#define SEQ 1024
#define KCHUNK 64
#define LDS_STRIDE 68   // 64 + 4 pad floats -> conflict-free strided re-read

// ---------------------------------------------------------------------------
// Kernel 1: token_logits[b,s] = dot(seq[b,s,:], w) + bias  via f32 WMMA.
// 8 waves / block, 16 tokens / wave, K streamed in chunks of 64 through LDS.
// ---------------------------------------------------------------------------
__global__ __launch_bounds__(256) void token_logits_wmma_kernel(
    const float* __restrict__ seq,      // [B*S, H]
    const float* __restrict__ w,        // [H]
    const float* __restrict__ bias_p,   // [1]
    float* __restrict__ token_logits)   // [B*S]
{
    __shared__ float w_lds[HIDDEN];
    __shared__ float tile[8][16 * LDS_STRIDE];   // per-wave 16x64 (padded) tile

    const int tid  = threadIdx.x;
    const int wave = tid >> 5;
    const int lane = tid & 31;

    // stage weights once (broadcast reads later -> no bank conflicts)
    for (int i = tid; i < HIDDEN; i += 256) w_lds[i] = w[i];

    const float bias = bias_p[0];
    const int tokenBase = blockIdx.x * 128 + wave * 16;
    const float* base = seq + (size_t)tokenBase * HIDDEN;

    // loader roles: lane pair (2m, 2m+1) covers row m's 64-float chunk
    const int load_m   = lane >> 1;
    const int load_off = (lane & 1) * 32;

    // WMMA A-fragment roles (ISA 16x4 f32 A layout):
    // lanes 0-15: M=lane, K={0,1}; lanes 16-31: M=lane-16, K={2,3}
    const int a_m = lane & 15;
    const int a_k = (lane >> 4) * 2;

    float* mytile = &tile[wave][0];
    const float* arow = mytile + a_m * LDS_STRIDE;

    v8f c = {0.f, 0.f, 0.f, 0.f, 0.f, 0.f, 0.f, 0.f};

    __syncthreads();

    for (int k0 = 0; k0 < HIDDEN; k0 += KCHUNK) {
        const float* src = base + (size_t)load_m * HIDDEN + k0 + load_off;
        float*       dst = mytile + load_m * LDS_STRIDE + load_off;

        if (k0 + KCHUNK < HIDDEN) {
            __builtin_prefetch(src + KCHUNK, 0, 1);   // global_prefetch_b8
        }

        #pragma unroll
        for (int j = 0; j < 32; j += 4) {
            v4f v = *(const v4f*)(src + j);           // global_load_b128
            *(v4f*)(dst + j) = v;                     // ds_store_b128
        }

        __syncthreads();  // make the staged tile visible across lanes/waves

        #pragma unroll
        for (int kk = 0; kk < KCHUNK; kk += 4) {
            v2f a;
            a.x = arow[kk + a_k];
            a.y = arow[kk + a_k + 1];
            // B[k][n] = w[k0+kk+k] for all n (broadcast weight column):
            // VGPR0: lanes0-15 K=0, lanes16-31 K=1 ; VGPR1: K=2 / K=3
            v2f b;
            b.x = (lane < 16) ? w_lds[k0 + kk + 0] : w_lds[k0 + kk + 1];
            b.y = (lane < 16) ? w_lds[k0 + kk + 2] : w_lds[k0 + kk + 3];
            c = __builtin_amdgcn_wmma_f32_16x16x4_f32(
                    /*neg_a=*/false, a, /*neg_b=*/false, b,
                    /*c_mod=*/(short)0, c, /*reuse_a=*/false, /*reuse_b=*/false);
        }

        __syncthreads();  // WAR: tile re-staged next iteration
    }

    // D layout (16x16 f32 C/D): lane 0 holds M=0..7 (N=0) in c[0..7],
    // lane 16 holds M=8..15. All N columns identical by construction.
    if ((lane & 15) == 0) {
        const int mbase = (lane >> 4) * 8;
        #pragma unroll
        for (int r = 0; r < 8; ++r) {
            token_logits[tokenBase + mbase + r] = c[r] + bias;
        }
    }
}

// ---------------------------------------------------------------------------
// Kernel 2: per-batch segment-mean over cells + masked column reduction.
// One block per batch element; LDS atomics for cell sums/counts.
// ---------------------------------------------------------------------------
__global__ __launch_bounds__(256) void column_logits_kernel(
    const float* __restrict__ token_logits, // [B, S]
    const float* __restrict__ cell_mask,    // [B, NUM_CELLS]
    const int*   __restrict__ row_ids,      // [B, S]
    const int*   __restrict__ col_ids,      // [B, S]
    const int*   __restrict__ allow_empty,  // [1]
    float* __restrict__ out)                // [B, NUM_COLS]
{
    __shared__ float s_sum[NUM_CELLS];
    __shared__ int   s_cnt[NUM_CELLS];

    const int b   = blockIdx.x;
    const int tid = threadIdx.x;

    for (int i = tid; i < NUM_CELLS; i += 256) { s_sum[i] = 0.f; s_cnt[i] = 0; }
    __syncthreads();

    const float* tl = token_logits + b * SEQ;
    const int*   rr = row_ids + b * SEQ;
    const int*   cc = col_ids + b * SEQ;

    for (int s = tid; s < SEQ; s += 256) {
        const int cell = rr[s] * NUM_COLS + cc[s];
        atomicAdd(&s_sum[cell], tl[s]);   // ds_add_f32
        atomicAdd(&s_cnt[cell], 1);       // ds_add_u32
    }
    __syncthreads();

    // deterministic column walk: 32 threads, 64 rows each
    if (tid < NUM_COLS) {
        const float* mask = cell_mask + b * NUM_CELLS;
        float acc = 0.f, mcnt = 0.f;
        #pragma unroll 4
        for (int r = 0; r < NUM_ROWS; ++r) {
            const int cell = r * NUM_COLS + tid;
            const int n = s_cnt[cell];
            const float mean = (n > 0) ? (s_sum[cell] / (float)n) : 0.f;
            const float m = mask[cell];
            acc  += mean * m;
            mcnt += m;
        }
        float logit = acc / (mcnt + 1e-10f);
        if (mcnt < 0.5f && tid != 0) logit += -10000.0f;   // padding columns
        if (allow_empty[0] == 0 && tid == 0) logit += -10000.0f;
        out[b * NUM_COLS + tid] = logit;
    }
}

// ---------------------------------------------------------------------------
extern "C" void kernel_launch(void* const* d_in, const int* in_sizes, int n_in,
                              void* d_out, int out_size, void* d_ws, size_t ws_size,
                              hipStream_t stream) {
    const float* seq   = (const float*)d_in[0];  // [B,S,H]
    const float* mask  = (const float*)d_in[1];  // [B,NUM_CELLS]
    const float* w     = (const float*)d_in[2];  // [H]
    const float* bias  = (const float*)d_in[3];  // [1]
    const int*   rows  = (const int*)d_in[4];    // [B,S]
    const int*   cols  = (const int*)d_in[5];    // [B,S]
    const int*   allow = (const int*)d_in[6];    // [1]
    float* out = (float*)d_out;

    float* token_logits = (float*)d_ws;          // B*S floats = 256 KB

    const int total_tokens = BATCH * SEQ;        // 65536
    token_logits_wmma_kernel<<<total_tokens / 128, 256, 0, stream>>>(
        seq, w, bias, token_logits);
    column_logits_kernel<<<BATCH, 256, 0, stream>>>(
        token_logits, mask, rows, cols, allow, out);
}